// SphericalHarmonicsLayer_70497593196803
// MI455X (gfx1250) — compile-verified
//
#include <hip/hip_runtime.h>
#include <math.h>

// ---------------------------------------------------------------------------
// SphericalHarmonicsLayer for MI455X (gfx1250, wave32)
//
//  out[n, h*17 + j] , j=0..15 SH terms (L=3), j=16 = time channel
//  theta = x[n,0]*w_theta[h] + b_theta[h]   (same for phi / time)
//
//  The three 1->64 linears are one rank-4 GEMM:  [N,4] x [4, 3*64]
//  with A row = (x0, x1, x2, 1).  We feed it to V_WMMA_F32_16X16X4_F32,
//  one 16-sample tile per wave, 16 h-columns per WMMA, 3 WMMAs (theta/phi/time)
//  per column group.  SH evaluation + stores are per-lane VALU.
// ---------------------------------------------------------------------------

typedef __attribute__((ext_vector_type(2))) float v2f;
typedef __attribute__((ext_vector_type(8))) float v8f;

#define HIDDEN 64
#define OUTC   17          // 16 SH terms + time
#define TILE_N 16          // samples per wave tile
#define WAVES_PER_BLOCK 8  // 256 threads

struct ShCoef { float c[16]; };  // Klm (with sqrt(2) folded in for m != 0)

__global__ __launch_bounds__(256)
void sh_wmma_kernel(const float* __restrict__ x,        // [N,3]
                    const float* __restrict__ w_theta, const float* __restrict__ b_theta,
                    const float* __restrict__ w_phi,   const float* __restrict__ b_phi,
                    const float* __restrict__ w_time,  const float* __restrict__ b_time,
                    float* __restrict__ out,           // [N, 64*17]
                    ShCoef K, int nsamples)
{
    const int lane = threadIdx.x & 31;
    const int wave = threadIdx.x >> 5;
    const int tile = blockIdx.x * WAVES_PER_BLOCK + wave;
    const int n0   = tile * TILE_N;
    if (n0 >= nsamples) return;            // uniform per wave -> full-EXEC inside

    const int hi  = lane >> 4;             // 0: lanes 0..15, 1: lanes 16..31
    const int idx = lane & 15;

    // ----- A matrix (16x4 f32): row m = (x0, x1, x2, 1) for sample n0+m -----
    // ISA layout: lanes 0-15 -> {K0,K1}, lanes 16-31 -> {K2,K3}
    int nA = n0 + idx; if (nA > nsamples - 1) nA = nsamples - 1;   // clamp (no divergence)
    v2f A;
    A.x = x[nA * 3 + (hi ? 2 : 0)];        // K0 = x0            | K2 = x2
    A.y = hi ? 1.0f : x[nA * 3 + 1];       // K1 = x1            | K3 = 1

    #pragma unroll
    for (int g = 0; g < HIDDEN / 16; ++g) {
        const int h = g * 16 + idx;        // output column (N dim of WMMA)
        const float wt = w_theta[h], bt = b_theta[h];
        const float wp = w_phi[h],   bp = b_phi[h];
        const float wm = w_time[h],  bm = b_time[h];

        // B (4x16 f32, column-major load): lane = column, VGPR0={K0|K2}, VGPR1={K1|K3}
        v2f Bt, Bp, Bm;
        Bt.x = hi ? 0.0f : wt;   Bt.y = hi ? bt : 0.0f;   // theta: row0=w, row3=b
        Bp.x = 0.0f;             Bp.y = hi ? bp : wp;     // phi:   row1=w, row3=b
        Bm.x = hi ? wm : 0.0f;   Bm.y = hi ? bm : 0.0f;   // time:  row2=w, row3=b

        v8f z = {0.f,0.f,0.f,0.f,0.f,0.f,0.f,0.f};
        v8f Dth = __builtin_amdgcn_wmma_f32_16x16x4_f32(false, A, false, Bt, (short)0, z, false, false);
        v8f Dph = __builtin_amdgcn_wmma_f32_16x16x4_f32(false, A, false, Bp, (short)0, z, false, false);
        v8f Dtm = __builtin_amdgcn_wmma_f32_16x16x4_f32(false, A, false, Bm, (short)0, z, false, false);

        // D layout: VGPR k, lanes 0-15 -> (M=k, N=lane); lanes 16-31 -> (M=k+8, N=lane-16)
        #pragma unroll
        for (int k = 0; k < 8; ++k) {
            const int nn = n0 + k + 8 * hi;
            if (nn >= nsamples) continue;

            const float theta = Dth[k];
            const float phi   = Dph[k];
            const float tval  = Dtm[k];

            // --- associated Legendre P_l^m(cos theta), Condon-Shortley, L=3 ---
            const float ct = __cosf(theta);
            const float st = sqrtf(fmaxf(1.0f - ct * ct, 0.0f));
            float sp, cp;  __sincosf(phi, &sp, &cp);
            const float c2 = cp * cp - sp * sp, s2 = 2.0f * cp * sp;
            const float c3 = c2 * cp - s2 * sp, s3 = s2 * cp + c2 * sp;

            const float P10 = ct;
            const float P11 = -st;
            const float P20 = 0.5f * (3.0f * ct * ct - 1.0f);
            const float P21 = -3.0f * ct * st;
            const float P22 = 3.0f * st * st;
            const float P30 = (5.0f * ct * P20 - 2.0f * P10) * (1.0f / 3.0f);
            const float P31 = (5.0f * ct * P21 - 3.0f * P11) * 0.5f;
            const float P32 = 5.0f * ct * P22;
            const float P33 = -5.0f * st * P22;

            float* o = out + (size_t)nn * (HIDDEN * OUTC) + h * OUTC;
            // output never re-read & exceeds L2 -> non-temporal stores
            __builtin_nontemporal_store(K.c[0],               o + 0);
            __builtin_nontemporal_store(K.c[1]  * P11 * sp,   o + 1);
            __builtin_nontemporal_store(K.c[2]  * P10,        o + 2);
            __builtin_nontemporal_store(K.c[3]  * P11 * cp,   o + 3);
            __builtin_nontemporal_store(K.c[4]  * P22 * s2,   o + 4);
            __builtin_nontemporal_store(K.c[5]  * P21 * sp,   o + 5);
            __builtin_nontemporal_store(K.c[6]  * P20,        o + 6);
            __builtin_nontemporal_store(K.c[7]  * P21 * cp,   o + 7);
            __builtin_nontemporal_store(K.c[8]  * P22 * c2,   o + 8);
            __builtin_nontemporal_store(K.c[9]  * P33 * s3,   o + 9);
            __builtin_nontemporal_store(K.c[10] * P32 * s2,   o + 10);
            __builtin_nontemporal_store(K.c[11] * P31 * sp,   o + 11);
            __builtin_nontemporal_store(K.c[12] * P30,        o + 12);
            __builtin_nontemporal_store(K.c[13] * P31 * cp,   o + 13);
            __builtin_nontemporal_store(K.c[14] * P32 * c2,   o + 14);
            __builtin_nontemporal_store(K.c[15] * P33 * c3,   o + 15);
            __builtin_nontemporal_store(tval,                 o + 16);
        }
    }
}

static double fact_d(int k) { double f = 1.0; for (int i = 2; i <= k; ++i) f *= (double)i; return f; }

extern "C" void kernel_launch(void* const* d_in, const int* in_sizes, int n_in,
                              void* d_out, int out_size, void* d_ws, size_t ws_size,
                              hipStream_t stream) {
    (void)n_in; (void)out_size; (void)d_ws; (void)ws_size;

    const float* x       = (const float*)d_in[0];
    const float* w_theta = (const float*)d_in[1];
    const float* b_theta = (const float*)d_in[2];
    const float* w_phi   = (const float*)d_in[3];
    const float* b_phi   = (const float*)d_in[4];
    const float* w_time  = (const float*)d_in[5];
    const float* b_time  = (const float*)d_in[6];
    float* out = (float*)d_out;

    const int nsamples = in_sizes[0] / 3;   // B=1

    // Klm exactly as the reference computes it (signed m inside the factorials),
    // with sqrt(2) folded in for m != 0.  Host double math, passed by value.
    ShCoef K;
    {
        const double four_pi = 4.0 * M_PI;
        const double sqrt2 = sqrt(2.0);
        int j = 0;
        for (int l = 0; l <= 3; ++l) {
            for (int m = -l; m <= l; ++m) {
                double klm = sqrt((2.0 * l + 1.0) * fact_d(l - m) / (four_pi * fact_d(l + m)));
                if (m != 0) klm *= sqrt2;
                K.c[j++] = (float)klm;
            }
        }
    }

    const int tiles  = (nsamples + TILE_N - 1) / TILE_N;
    const int blocks = (tiles + WAVES_PER_BLOCK - 1) / WAVES_PER_BLOCK;

    sh_wmma_kernel<<<blocks, 256, 0, stream>>>(x, w_theta, b_theta, w_phi, b_phi,
                                               w_time, b_time, out, K, nsamples);
}